// GRURippleNet_7447473291747
// MI455X (gfx1250) — compile-verified
//
#include <hip/hip_runtime.h>
#include <math.h>

#define DIM   64
#define NHOP  2
#define NMEM  32
#define BATCH 512
#define NREL  32
#define NTOT  (NHOP*BATCH*NMEM)   // 32768
#define KGE_W 0.01f
#define L2_W  1e-7f
#define EPSV  1e-7f

typedef float v2f __attribute__((ext_vector_type(2)));
typedef float v8f __attribute__((ext_vector_type(8)));

__device__ __forceinline__ float sigmoidf_(float x) { return 1.f / (1.f + expf(-x)); }

// ---------------------------------------------------------------------------
// 1) per-relation squared norm:  relnorm[r] = sum(relation_emb[r]^2)
// ---------------------------------------------------------------------------
__global__ __launch_bounds__(256) void relnorm_kernel(const float* __restrict__ rel,
                                                      float* __restrict__ relnorm) {
    int r = blockIdx.x, tid = threadIdx.x;
    float acc = 0.f;
    for (int i = tid; i < DIM * DIM; i += 256) {
        float v = rel[r * DIM * DIM + i];
        acc += v * v;
    }
    for (int o = 16; o >= 1; o >>= 1) acc += __shfl_xor(acc, o, 32);
    __shared__ float red[8];
    if ((tid & 31) == 0) red[tid >> 5] = acc;
    __syncthreads();
    if (tid == 0) {
        float s = 0.f;
        for (int w = 0; w < 8; ++w) s += red[w];
        relnorm[r] = s;
    }
}

// ---------------------------------------------------------------------------
// 2) stable group-by-relation (deterministic counting sort; one wave32)
// ---------------------------------------------------------------------------
__global__ __launch_bounds__(32) void group_kernel(const int* __restrict__ mem_r,
                                                   int* __restrict__ counts,
                                                   int* __restrict__ offsets,
                                                   int* __restrict__ perm) {
    int r = threadIdx.x;  // one thread per relation, wave32
    int cnt = 0;
    for (int e = 0; e < NTOT; ++e) cnt += (mem_r[e] == r) ? 1 : 0;
    counts[r] = cnt;
    // inclusive scan within the wave -> exclusive offsets
    int inc = cnt;
    for (int d = 1; d < 32; d <<= 1) {
        int v = __shfl_up(inc, d, 32);
        if (r >= d) inc += v;
    }
    int excl = inc - cnt;
    offsets[r] = excl;
    if (r == 31) offsets[32] = excl + cnt;  // == NTOT
    int pos = excl;
    for (int e = 0; e < NTOT; ++e)
        if (mem_r[e] == r) perm[pos++] = e;  // stable order
}

// ---------------------------------------------------------------------------
// 3) item embedding gather
// ---------------------------------------------------------------------------
__global__ __launch_bounds__(64) void item_gather_kernel(const float* __restrict__ entity_emb,
                                                         const int* __restrict__ items,
                                                         float* __restrict__ item0) {
    int b = blockIdx.x, d = threadIdx.x;
    item0[b * DIM + d] = entity_emb[(size_t)items[b] * DIM + d];
}

// ---------------------------------------------------------------------------
// 4) grouped relation GEMM with V_WMMA_F32_16X16X4_F32:
//    for each relation r:   Rh = R_r @ H,   Rt = R_r @ T,  hRt = coldot(Rt,H)
//    fused: L2(h)+L2(t) partials and sum(sigmoid(hRt)) partials per block.
// ---------------------------------------------------------------------------
__global__ __launch_bounds__(128) void rgemm_kernel(const float* __restrict__ entity_emb,
                                                    const float* __restrict__ relation_emb,
                                                    const int* __restrict__ mem_h,
                                                    const int* __restrict__ mem_t,
                                                    const int* __restrict__ perm,
                                                    const int* __restrict__ offsets,
                                                    float* __restrict__ Rh,
                                                    float* __restrict__ l2part,
                                                    float* __restrict__ kgepart) {
    __shared__ float Rlds[DIM * DIM];      // 16 KB, relation matrix (row major)
    __shared__ float Hlds[4][DIM * 16];    // per-wave H tile, [k][col]
    __shared__ float Tlds[4][DIM * 16];    // per-wave T tile, [k][col]
    __shared__ int   Echunk[4][16];
    __shared__ float redA[4], redB[4];

    const int r = blockIdx.x;
    const int tid = threadIdx.x;
    const int lane = tid & 31;
    const int wave = tid >> 5;

    for (int i = tid; i < DIM * DIM; i += 128) Rlds[i] = relation_emb[r * DIM * DIM + i];
    __syncthreads();

    const int beg = offsets[r];
    const int Nr = offsets[r + 1] - beg;

    float l2acc = 0.f, kgeacc = 0.f;
    float* H = Hlds[wave];
    float* T = Tlds[wave];
    int*   E = Echunk[wave];

    // chunk loop: uniform per wave (no block barriers inside)
    for (int cb = wave * 16; cb < Nr; cb += 4 * 16) {
        // ---- stage 16 columns of H and T (k-major [64][16] in LDS) ----
        for (int s = 0; s < 16; ++s) {
            int gidx = cb + s;
            int e = (gidx < Nr) ? perm[beg + gidx] : -1;
            if (lane == 0) E[s] = e;
            float hv0 = 0.f, hv1 = 0.f, tv0 = 0.f, tv1 = 0.f;
            if (e >= 0) {
                size_t hb = (size_t)mem_h[e] * DIM;
                size_t tb = (size_t)mem_t[e] * DIM;
                hv0 = entity_emb[hb + lane];
                hv1 = entity_emb[hb + 32 + lane];
                tv0 = entity_emb[tb + lane];
                tv1 = entity_emb[tb + 32 + lane];
            }
            H[lane * 16 + s]        = hv0;
            H[(lane + 32) * 16 + s] = hv1;
            T[lane * 16 + s]        = tv0;
            T[(lane + 32) * 16 + s] = tv1;
            l2acc += hv0 * hv0 + hv1 * hv1 + tv0 * tv0 + tv1 * tv1;
        }
        // wave32 executes in lockstep; only need LDS write completion
        asm volatile("s_wait_dscnt 0" ::: "memory");

        // ---- WMMA compute: 4 row-tiles of 16, K accumulated 4 at a time ----
        const int col  = lane & 15;
        const int ksel = (lane < 16) ? 0 : 2;  // A/B: vgpr0 holds K=0/2, vgpr1 K=1/3
        float hrt = 0.f;
        for (int mt = 0; mt < 4; ++mt) {
            v8f c1 = {};  // Rh tile
            v8f c2 = {};  // Rt tile
            const int arow = mt * 16 + col;
            for (int kk = 0; kk < 16; ++kk) {
                const int kb = kk * 4 + ksel;
                v2f a, bh, bt;
                a.x  = Rlds[arow * DIM + kb];
                a.y  = Rlds[arow * DIM + kb + 1];
                bh.x = H[kb * 16 + col];
                bh.y = H[(kb + 1) * 16 + col];
                bt.x = T[kb * 16 + col];
                bt.y = T[(kb + 1) * 16 + col];
                c1 = __builtin_amdgcn_wmma_f32_16x16x4_f32(false, a, false, bh,
                                                           (short)0, c1, false, false);
                c2 = __builtin_amdgcn_wmma_f32_16x16x4_f32(false, a, false, bt,
                                                           (short)0, c2, false, false);
            }
            // scatter Rh, accumulate hRt partial (C layout: vgpr d -> M=d / M=d+8)
            const int e = E[col];
            const int rbase = mt * 16 + ((lane < 16) ? 0 : 8);
#pragma unroll
            for (int d = 0; d < 8; ++d) {
                const int row = rbase + d;
                if (e >= 0) Rh[(size_t)e * DIM + row] = c1[d];
                hrt += c2[d] * H[row * 16 + col];
            }
        }
        hrt += __shfl_xor(hrt, 16, 32);  // combine row halves (lane l and l+16 share a column)
        if (lane < 16) {
            int e = E[lane];
            if (e >= 0) kgeacc += sigmoidf_(hrt);
        }
    }

    // deterministic block reduction of the two partials
    for (int o = 16; o >= 1; o >>= 1) {
        l2acc  += __shfl_xor(l2acc, o, 32);
        kgeacc += __shfl_xor(kgeacc, o, 32);
    }
    if (lane == 0) { redA[wave] = l2acc; redB[wave] = kgeacc; }
    __syncthreads();
    if (tid == 0) {
        float a = 0.f, b = 0.f;
        for (int w = 0; w < 4; ++w) { a += redA[w]; b += redB[w]; }
        l2part[r] = a;
        kgepart[r] = b;
    }
}

// ---------------------------------------------------------------------------
// 5) serial GRU scan over the batch axis (PyTorch 2D-input semantics).
//    Gate order r,z,n. One block, 192 threads (one per gate row).
// ---------------------------------------------------------------------------
__global__ __launch_bounds__(192) void gru_kernel(const float* __restrict__ X,
                                                  float* __restrict__ Y,
                                                  float* __restrict__ hstate,
                                                  const float* __restrict__ wih,
                                                  const float* __restrict__ whh,
                                                  const float* __restrict__ bih,
                                                  const float* __restrict__ bhh,
                                                  int init_zero) {
    __shared__ float h[DIM];
    __shared__ float x[DIM];
    __shared__ float gi[3 * DIM];
    __shared__ float gh[3 * DIM];
    const int tid = threadIdx.x;
    if (tid < DIM) h[tid] = init_zero ? 0.f : hstate[tid];
    __syncthreads();
    for (int b = 0; b < BATCH; ++b) {
        if (tid < DIM) x[tid] = X[b * DIM + tid];
        __syncthreads();
        float a = bih[tid], c = bhh[tid];
        const float* wi = wih + tid * DIM;
        const float* wh = whh + tid * DIM;
#pragma unroll 8
        for (int j = 0; j < DIM; ++j) {
            a = fmaf(wi[j], x[j], a);
            c = fmaf(wh[j], h[j], c);
        }
        gi[tid] = a;
        gh[tid] = c;
        __syncthreads();
        if (tid < DIM) {
            float rr = sigmoidf_(gi[tid] + gh[tid]);
            float zz = sigmoidf_(gi[DIM + tid] + gh[DIM + tid]);
            float nn = tanhf(gi[2 * DIM + tid] + rr * gh[2 * DIM + tid]);
            float hn = (1.f - zz) * nn + zz * h[tid];
            h[tid] = hn;
            Y[b * DIM + tid] = hn;
        }
        __syncthreads();
    }
    if (tid < DIM) hstate[tid] = h[tid];
}

// ---------------------------------------------------------------------------
// 6) attention per batch element, both hops fused:
//    logits_m = Rh[b,m] . item(hop) ; softmax over M=32 in exactly one wave32;
//    o_sum[b] = sum_hops sum_m p_m * t_emb[b,m]
// ---------------------------------------------------------------------------
__global__ __launch_bounds__(64) void attn_kernel(const float* __restrict__ Rh,
                                                  const float* __restrict__ item0,
                                                  const float* __restrict__ item1,
                                                  const float* __restrict__ entity_emb,
                                                  const int* __restrict__ mem_t,
                                                  float* __restrict__ o_sum) {
    const int b = blockIdx.x, tid = threadIdx.x;
    __shared__ float p[NMEM];
    __shared__ float itemv[DIM];
    float oacc = 0.f;
    for (int hop = 0; hop < NHOP; ++hop) {
        const float* item = hop ? item1 : item0;
        itemv[tid] = item[b * DIM + tid];
        __syncthreads();
        if (tid < NMEM) {  // wave 0 only (wave32)
            const size_t e = (size_t)hop * BATCH * NMEM + (size_t)b * NMEM + tid;
            float lg = 0.f;
            for (int d = 0; d < DIM; ++d) lg += Rh[e * DIM + d] * itemv[d];
            float mx = lg;
            for (int o = 16; o >= 1; o >>= 1) mx = fmaxf(mx, __shfl_xor(mx, o, 32));
            float ex = expf(lg - mx);
            float sm = ex;
            for (int o = 16; o >= 1; o >>= 1) sm += __shfl_xor(sm, o, 32);
            p[tid] = ex / sm;
        }
        __syncthreads();
        float od = 0.f;
        const size_t ebase = (size_t)hop * BATCH * NMEM + (size_t)b * NMEM;
        for (int m = 0; m < NMEM; ++m)
            od = fmaf(p[m], entity_emb[(size_t)mem_t[ebase + m] * DIM + tid], od);
        oacc += od;
        __syncthreads();
    }
    o_sum[b * DIM + tid] = oacc;
}

// ---------------------------------------------------------------------------
// 7) scores + BCE partials
// ---------------------------------------------------------------------------
__global__ __launch_bounds__(64) void score_kernel(const float* __restrict__ item2,
                                                   const float* __restrict__ o_sum,
                                                   const int* __restrict__ labels,
                                                   float* __restrict__ out_scores,
                                                   float* __restrict__ bcepart) {
    const int b = blockIdx.x, tid = threadIdx.x;
    float v = item2[b * DIM + tid] * o_sum[b * DIM + tid];
    for (int o = 16; o >= 1; o >>= 1) v += __shfl_xor(v, o, 32);
    __shared__ float red[2];
    if ((tid & 31) == 0) red[tid >> 5] = v;
    __syncthreads();
    if (tid == 0) {
        float dot = red[0] + red[1];
        float s = sigmoidf_(dot);
        out_scores[b] = s;
        float sc = fminf(fmaxf(s, EPSV), 1.f - EPSV);
        float lbl = (float)labels[b];
        bcepart[b] = -(lbl * logf(sc) + (1.f - lbl) * log1pf(-sc));
    }
}

// ---------------------------------------------------------------------------
// 8) scalar finalize (single thread -> fully deterministic)
// ---------------------------------------------------------------------------
__global__ __launch_bounds__(1) void finalize_kernel(const float* __restrict__ bcepart,
                                                     const float* __restrict__ kgepart,
                                                     const float* __restrict__ l2part,
                                                     const int* __restrict__ counts,
                                                     const float* __restrict__ relnorm,
                                                     float* __restrict__ out4) {
    float bce = 0.f;
    for (int i = 0; i < BATCH; ++i) bce += bcepart[i];
    float base_loss = bce / (float)BATCH;

    float sig = 0.f, l2e = 0.f, l2r = 0.f;
    for (int r = 0; r < NREL; ++r) {
        sig += kgepart[r];
        l2e += l2part[r];
        l2r += (float)counts[r] * relnorm[r];
    }
    float kge_loss = -KGE_W * (sig / (float)(BATCH * NMEM));
    float l2_loss  = L2_W * (l2e + l2r);
    out4[0] = base_loss;
    out4[1] = kge_loss;
    out4[2] = l2_loss;
    out4[3] = base_loss + kge_loss + l2_loss;
}

// ---------------------------------------------------------------------------
extern "C" void kernel_launch(void* const* d_in, const int* in_sizes, int n_in,
                              void* d_out, int out_size, void* d_ws, size_t ws_size,
                              hipStream_t stream) {
    const int*   items        = (const int*)d_in[0];
    const int*   labels       = (const int*)d_in[1];
    const int*   mem_h        = (const int*)d_in[2];   // [2,512,32] flat
    const int*   mem_r        = (const int*)d_in[3];
    const int*   mem_t        = (const int*)d_in[4];
    const float* entity_emb   = (const float*)d_in[5];
    const float* relation_emb = (const float*)d_in[6];
    const float* wih          = (const float*)d_in[7];
    const float* whh          = (const float*)d_in[8];
    const float* bih          = (const float*)d_in[9];
    const float* bhh          = (const float*)d_in[10];
    float* out = (float*)d_out;  // [512 scores | base | kge | l2 | loss]

    char* ws = (char*)d_ws;
    auto alloc = [&](size_t bytes) -> char* {
        char* p = ws;
        ws += (bytes + 255) & ~(size_t)255;
        return p;
    };
    int*   perm    = (int*)alloc((size_t)NTOT * 4);
    int*   counts  = (int*)alloc(32 * 4);
    int*   offsets = (int*)alloc(33 * 4);
    float* Rh      = (float*)alloc((size_t)NTOT * DIM * 4);   // 8.4 MB
    float* item0   = (float*)alloc((size_t)BATCH * DIM * 4);
    float* item1   = (float*)alloc((size_t)BATCH * DIM * 4);
    float* item2   = (float*)alloc((size_t)BATCH * DIM * 4);
    float* o_sum   = (float*)alloc((size_t)BATCH * DIM * 4);
    float* hstate  = (float*)alloc(DIM * 4);
    float* relnorm = (float*)alloc(NREL * 4);
    float* l2part  = (float*)alloc(NREL * 4);
    float* kgepart = (float*)alloc(NREL * 4);
    float* bcepart = (float*)alloc(BATCH * 4);

    relnorm_kernel<<<NREL, 256, 0, stream>>>(relation_emb, relnorm);
    group_kernel<<<1, 32, 0, stream>>>(mem_r, counts, offsets, perm);
    item_gather_kernel<<<BATCH, 64, 0, stream>>>(entity_emb, items, item0);
    rgemm_kernel<<<NREL, 128, 0, stream>>>(entity_emb, relation_emb, mem_h, mem_t,
                                           perm, offsets, Rh, l2part, kgepart);
    gru_kernel<<<1, 192, 0, stream>>>(item0, item1, hstate, wih, whh, bih, bhh, 1);
    attn_kernel<<<BATCH, 64, 0, stream>>>(Rh, item0, item1, entity_emb, mem_t, o_sum);
    gru_kernel<<<1, 192, 0, stream>>>(item1, item2, hstate, wih, whh, bih, bhh, 0);
    score_kernel<<<BATCH, 64, 0, stream>>>(item2, o_sum, labels, out, bcepart);
    finalize_kernel<<<1, 1, 0, stream>>>(bcepart, kgepart, l2part, counts, relnorm, out + BATCH);
}